// NerfModel_59253368816339
// MI455X (gfx1250) — compile-verified
//
#include <hip/hip_runtime.h>
#include <math.h>

#define RES   192
#define SCALE 1.5f
#define NCH   28
#define CPAD  32    // pad 28 channels -> 32 floats = exactly one 128B cacheline/voxel
#define TVOX  512   // voxels per TDM tile: 28*512*4B = 56KB LDS

typedef unsigned int v4u __attribute__((ext_vector_type(4)));
typedef int          v8i __attribute__((ext_vector_type(8)));
typedef int          v4i __attribute__((ext_vector_type(4)));

// ---------------------------------------------------------------------------
// Pass 1 (TDM path): transpose [C=28, D,H,W] -> [D,H,W, C=32].
// One wave per block issues a single tensor_load_to_lds moving a 2D tile
// (tile_dim0 = 512 contiguous voxels, tile_dim1 = 28 channel rows,
//  tensor_dim0_stride = 192^3 elements) into LDS as [28][512] f32, waits on
// TENSORcnt, then all waves read channel-major from LDS (conflict-free) and
// write padded 128B voxel lines as coalesced b128 stores.
// D# bitfields built per CDNA5 ISA.
// ---------------------------------------------------------------------------
__global__ __launch_bounds__(256) void transpose_grid_tdm_kernel(
    const float* __restrict__ g, float* __restrict__ gp)
{
    __shared__ float tile[NCH * TVOX];

    const long long V  = (long long)RES * RES * RES;
    const long long v0 = (long long)blockIdx.x * TVOX;

    if (threadIdx.x < 32) {  // wave 0 owns the DMA
        unsigned long long gaddr = (unsigned long long)(uintptr_t)(g + v0);
        unsigned int       laddr = (unsigned int)(uintptr_t)&tile[0]; // LDS byte offset

        // ---- D# group 0 (128b): count=1 | lds_addr | global_addr[56:0] | type=2
        v4u g0;
        g0.x = 1u;                                        // count=1, user load desc
        g0.y = laddr;                                     // lds_addr      [63:32]
        g0.z = (unsigned int)(gaddr & 0xFFFFFFFFull);     // global_addr   [95:64]
        g0.w = (unsigned int)((gaddr >> 32) & 0x01FFFFFFu)// global_addr   [120:96]
             | (2u << 30);                                // type=2 ("image")

        // ---- D# group 1 (256b)
        v8i g1;
        g1[0] = (int)(2u << 16);                          // data_size=2 -> 4 bytes
        g1[1] = (int)(((unsigned)TVOX & 0xFFFFu) << 16);  // tensor_dim0 lo16 @[63:48]
        g1[2] = (int)((((unsigned)TVOX >> 16) & 0xFFFFu)  // tensor_dim0 hi16 @[79:64]
             | (((unsigned)NCH & 0xFFFFu) << 16));        // tensor_dim1 lo16 @[95:80]
        g1[3] = (int)(((unsigned)NCH >> 16)               // tensor_dim1 hi16 @[111:96]
             | (((unsigned)TVOX & 0xFFFFu) << 16));       // tile_dim0        @[127:112]
        g1[4] = (int)((unsigned)NCH & 0xFFFFu);           // tile_dim1=28, tile_dim2=0
        g1[5] = (int)(unsigned int)(V & 0xFFFFFFFFll);    // tensor_dim0_stride lo32
        g1[6] = (int)(unsigned int)((V >> 32) & 0xFFFFll);// tensor_dim0_stride hi16; dim1_stride lo16=0
        g1[7] = 0;                                        // tensor_dim1_stride hi32=0

        v4i gz4 = {0, 0, 0, 0};                           // groups 2/3 unused (2D tensor)
        v8i gz8 = {0, 0, 0, 0, 0, 0, 0, 0};               // unused trailing group (clang-23 6-arg form)
        __builtin_amdgcn_tensor_load_to_lds(g0, g1, gz4, gz4, gz8, 0);
        __builtin_amdgcn_s_wait_tensorcnt(0);
    }
    __syncthreads();

    // Each thread finalizes 2 voxels (t and t+256): gather 28 channels from
    // LDS (fixed c -> lane-consecutive addrs, bank-conflict-free), emit one
    // padded 128B line per voxel as 8 x b128 coalesced stores.
#pragma unroll
    for (int k = 0; k < 2; ++k) {
        int v = (int)threadIdx.x + k * 256;
        float vals[NCH];
#pragma unroll
        for (int c = 0; c < NCH; ++c) vals[c] = tile[c * TVOX + v];
        float4* dst = (float4*)(gp + ((v0 + v) << 5));
#pragma unroll
        for (int q = 0; q < 7; ++q)
            dst[q] = make_float4(vals[4*q], vals[4*q+1], vals[4*q+2], vals[4*q+3]);
        dst[7] = make_float4(0.f, 0.f, 0.f, 0.f);         // pad lane (never read)
    }
}

// ---------------------------------------------------------------------------
// Accumulate one corner: 7 x global_load_b128 from one cacheline, FMA'd with
// the trilinear weight into the 28 channel accumulators.
// ---------------------------------------------------------------------------
__device__ __forceinline__ void acc_corner(const float* __restrict__ p,
                                           float w, float acc[NCH])
{
    const float4* __restrict__ s = (const float4*)p;
#pragma unroll
    for (int q = 0; q < 7; ++q) {
        float4 t = s[q];
        acc[4*q+0] = fmaf(w, t.x, acc[4*q+0]);
        acc[4*q+1] = fmaf(w, t.y, acc[4*q+1]);
        acc[4*q+2] = fmaf(w, t.z, acc[4*q+2]);
        acc[4*q+3] = fmaf(w, t.w, acc[4*q+3]);
    }
}

// Shared tail: SH basis (deg 2) + output writes. out layout: [sigma(N) | c(3N)]
__device__ __forceinline__ void finish_point(const float acc[NCH],
                                             const float* __restrict__ dir,
                                             float* __restrict__ out_sigma,
                                             float* __restrict__ out_c,
                                             int i)
{
    float dx = dir[3*i + 0], dy = dir[3*i + 1], dz = dir[3*i + 2];
    float b0 =  0.282095f;
    float b1 = -0.488603f * dy;
    float b2 =  0.488603f * dz;
    float b3 = -0.488603f * dx;
    float b4 =  1.092548f * dx * dy;
    float b5 = -1.092548f * dy * dz;
    float b6 =  0.315392f * (2.0f * dz * dz - dx * dx - dy * dy);
    float b7 = -1.092548f * dx * dz;
    float b8 =  0.546274f * (dx * dx - dy * dy);

    out_sigma[i] = fmaxf(acc[0], 0.0f);
#pragma unroll
    for (int j = 0; j < 3; ++j) {
        const float* k = acc + 1 + 9*j;
        float cj = b0*k[0];
        cj = fmaf(b1, k[1], cj); cj = fmaf(b2, k[2], cj);
        cj = fmaf(b3, k[3], cj); cj = fmaf(b4, k[4], cj);
        cj = fmaf(b5, k[5], cj); cj = fmaf(b6, k[6], cj);
        cj = fmaf(b7, k[7], cj); cj = fmaf(b8, k[8], cj);
        out_c[3*i + j] = cj;
    }
}

// Shared index/weight computation (align_corners=True trilinear).
struct TriIdx {
    int ix0, ix1, iy0, iy1, iz0, iz1;
    float w[8];  // bit2=z1, bit1=y1, bit0=x1
};
__device__ __forceinline__ TriIdx tri_setup(float x0, float x1, float x2)
{
    TriIdx t;
    const float s = 0.5f * (RES - 1);
    float px = (fminf(fmaxf(x0 * (1.0f/SCALE), -1.0f), 1.0f) + 1.0f) * s;
    float py = (fminf(fmaxf(x1 * (1.0f/SCALE), -1.0f), 1.0f) + 1.0f) * s;
    float pz = (fminf(fmaxf(x2 * (1.0f/SCALE), -1.0f), 1.0f) + 1.0f) * s;
    float fx = floorf(px), fy = floorf(py), fz = floorf(pz);
    t.ix0 = (int)fx; t.iy0 = (int)fy; t.iz0 = (int)fz;
    t.ix1 = min(t.ix0 + 1, RES - 1);
    t.iy1 = min(t.iy0 + 1, RES - 1);
    t.iz1 = min(t.iz0 + 1, RES - 1);
    float wx = px - fx, wy = py - fy, wz = pz - fz;
    float wx0 = 1.f - wx, wy0 = 1.f - wy, wz0 = 1.f - wz;
    t.w[0] = wz0*wy0*wx0; t.w[1] = wz0*wy0*wx;
    t.w[2] = wz0*wy *wx0; t.w[3] = wz0*wy *wx;
    t.w[4] = wz *wy0*wx0; t.w[5] = wz *wy0*wx;
    t.w[6] = wz *wy *wx0; t.w[7] = wz *wy *wx;
    return t;
}

// ---------------------------------------------------------------------------
// Pass 2 (fast path): gather from padded [D,H,W,32] layout.
// 8 cachelines per point instead of ~112.
// ---------------------------------------------------------------------------
__global__ __launch_bounds__(256) void nerf_eval_pad_kernel(
    const float* __restrict__ x, const float* __restrict__ dir,
    const float* __restrict__ gp,
    float* __restrict__ out_sigma, float* __restrict__ out_c, int n)
{
    int i = blockIdx.x * blockDim.x + threadIdx.x;
    if (i >= n) return;

    float x0 = x[3*i + 0], x1 = x[3*i + 1], x2 = x[3*i + 2];
    bool inb = (fabsf(x0) < SCALE) & (fabsf(x1) < SCALE) & (fabsf(x2) < SCALE);

    float acc[NCH];
#pragma unroll
    for (int c = 0; c < NCH; ++c) acc[c] = 0.0f;

    if (inb) {
        TriIdx t = tri_setup(x0, x1, x2);
        long long z0 = (long long)t.iz0 * RES, z1 = (long long)t.iz1 * RES;
        long long r00 = (z0 + t.iy0) * RES, r01 = (z0 + t.iy1) * RES;
        long long r10 = (z1 + t.iy0) * RES, r11 = (z1 + t.iy1) * RES;
        acc_corner(gp + ((r00 + t.ix0) << 5), t.w[0], acc);
        acc_corner(gp + ((r00 + t.ix1) << 5), t.w[1], acc);
        acc_corner(gp + ((r01 + t.ix0) << 5), t.w[2], acc);
        acc_corner(gp + ((r01 + t.ix1) << 5), t.w[3], acc);
        acc_corner(gp + ((r10 + t.ix0) << 5), t.w[4], acc);
        acc_corner(gp + ((r10 + t.ix1) << 5), t.w[5], acc);
        acc_corner(gp + ((r11 + t.ix0) << 5), t.w[6], acc);
        acc_corner(gp + ((r11 + t.ix1) << 5), t.w[7], acc);
    }
    finish_point(acc, dir, out_sigma, out_c, i);
}

// ---------------------------------------------------------------------------
// Fallback: direct gather from original [C,D,H,W] layout (only if d_ws cannot
// hold the 906MB padded grid). ~112 cachelines/point.
// ---------------------------------------------------------------------------
__global__ __launch_bounds__(256) void nerf_eval_direct_kernel(
    const float* __restrict__ x, const float* __restrict__ dir,
    const float* __restrict__ g,
    float* __restrict__ out_sigma, float* __restrict__ out_c, int n)
{
    int i = blockIdx.x * blockDim.x + threadIdx.x;
    if (i >= n) return;

    float x0 = x[3*i + 0], x1 = x[3*i + 1], x2 = x[3*i + 2];
    bool inb = (fabsf(x0) < SCALE) & (fabsf(x1) < SCALE) & (fabsf(x2) < SCALE);

    float acc[NCH];
#pragma unroll
    for (int c = 0; c < NCH; ++c) acc[c] = 0.0f;

    if (inb) {
        TriIdx t = tri_setup(x0, x1, x2);
        const long long V = (long long)RES * RES * RES;
        long long z0 = (long long)t.iz0 * RES, z1 = (long long)t.iz1 * RES;
        long long r00 = (z0 + t.iy0) * RES, r01 = (z0 + t.iy1) * RES;
        long long r10 = (z1 + t.iy0) * RES, r11 = (z1 + t.iy1) * RES;
        for (int c = 0; c < NCH; ++c) {
            const float* __restrict__ gc = g + (long long)c * V;
            float v = 0.0f;
            v = fmaf(t.w[0], gc[r00 + t.ix0], v);
            v = fmaf(t.w[1], gc[r00 + t.ix1], v);
            v = fmaf(t.w[2], gc[r01 + t.ix0], v);
            v = fmaf(t.w[3], gc[r01 + t.ix1], v);
            v = fmaf(t.w[4], gc[r10 + t.ix0], v);
            v = fmaf(t.w[5], gc[r10 + t.ix1], v);
            v = fmaf(t.w[6], gc[r11 + t.ix0], v);
            v = fmaf(t.w[7], gc[r11 + t.ix1], v);
            acc[c] = v;
        }
    }
    finish_point(acc, dir, out_sigma, out_c, i);
}

extern "C" void kernel_launch(void* const* d_in, const int* in_sizes, int n_in,
                              void* d_out, int out_size, void* d_ws, size_t ws_size,
                              hipStream_t stream)
{
    (void)n_in; (void)out_size;
    const float* x    = (const float*)d_in[0];   // [N,3] f32
    const float* dir  = (const float*)d_in[1];   // [N,3] f32
    const float* grid = (const float*)d_in[2];   // [28,192,192,192] f32

    const int n = in_sizes[0] / 3;
    float* out_sigma = (float*)d_out;            // N f32
    float* out_c     = (float*)d_out + n;        // 3N f32

    const long long nvox   = (long long)RES * RES * RES;
    const size_t    needed = (size_t)nvox * CPAD * sizeof(float);

    const int TB = 256;  // 8 wave32s per block
    if (ws_size >= needed) {
        float* gp = (float*)d_ws;
        unsigned tblocks = (unsigned)(nvox / TVOX);  // 192^3 % 512 == 0
        transpose_grid_tdm_kernel<<<tblocks, TB, 0, stream>>>(grid, gp);
        nerf_eval_pad_kernel<<<(n + TB - 1) / TB, TB, 0, stream>>>(
            x, dir, gp, out_sigma, out_c, n);
    } else {
        nerf_eval_direct_kernel<<<(n + TB - 1) / TB, TB, 0, stream>>>(
            x, dir, grid, out_sigma, out_c, n);
    }
}